// SCADeformableAttention_63402307223822
// MI455X (gfx1250) — compile-verified
//
#include <hip/hip_runtime.h>
#include <math.h>

#define BB 2
#define VV 3
#define GG 2
#define NHH 8
#define CC 128
#define CPG 64
#define CHH 16
#define HK 16
#define WK 128
#define NSS 2048
#define HI 64
#define WI 64
#define MQ 1024
#define RPE_H 63
#define RPE_W 255
#define SCALE_F 0.25f
#define OFR_F 5.0f
#define EPS_F 1e-5f

typedef __attribute__((ext_vector_type(2))) float v2f;
typedef __attribute__((ext_vector_type(8))) float v8f;

__device__ __forceinline__ v8f wmma4(v2f a, v2f b, v8f c) {
  // D(16x16 f32) = A(16x4 f32) * B(4x16 f32) + C
  return __builtin_amdgcn_wmma_f32_16x16x4_f32(false, a, false, b, (short)0, c, false, false);
}

__device__ __forceinline__ float wave_sum32(float v) {
#pragma unroll
  for (int m = 1; m <= 16; m <<= 1) v += __shfl_xor(v, m, 32);
  return v;
}
__device__ __forceinline__ float half_sum16(float v) {
#pragma unroll
  for (int m = 1; m <= 8; m <<= 1) v += __shfl_xor(v, m, 32);
  return v;
}
__device__ __forceinline__ float half_max16(float v) {
#pragma unroll
  for (int m = 1; m <= 8; m <<= 1) v = fmaxf(v, __shfl_xor(v, m, 32));
  return v;
}

// ---------------- offset net: one wave per (vi, bg, query pixel) -------------
__global__ void k_offset(const float* __restrict__ query,
                         const float* __restrict__ refp,
                         const float* __restrict__ w1, const float* __restrict__ b1,
                         const float* __restrict__ gln, const float* __restrict__ bln,
                         const float* __restrict__ w2, float* __restrict__ rwo) {
  int wg = blockIdx.x * (blockDim.x >> 5) + (threadIdx.x >> 5);
  int lane = threadIdx.x & 31;
  int pos = wg & (MQ - 1);
  int bg = (wg >> 10) & 3;
  int vi = wg >> 12;
  if (vi >= VV) return;
  int hq = pos >> 5, wq = pos & 31;
  int b = bg >> 1, g = bg & 1;

  float qa = query[(size_t)(b * CC + g * CPG + lane) * MQ + pos];
  float qb = query[(size_t)(b * CC + g * CPG + lane + 32) * MQ + pos];

  const float* w1v = w1 + vi * 256;
  const float* b1v = b1 + vi * 256;
  float h[8];
  float s = 0.f, s2 = 0.f;
#pragma unroll
  for (int j = 0; j < 8; ++j) {
    int c = (j < 4) ? lane : (lane + 32);
    int k = c * 4 + (j & 3);
    float hv = ((j < 4) ? qa : qb) * w1v[k] + b1v[k];
    h[j] = hv;
    s += hv;
    s2 += hv * hv;
  }
  s = wave_sum32(s);
  s2 = wave_sum32(s2);
  float mu = s * (1.f / 256.f);
  float var = s2 * (1.f / 256.f) - mu * mu;
  float rstd = rsqrtf(var + EPS_F);
  const float* glv = gln + vi * 256;
  const float* blv = bln + vi * 256;
  const float* w2v = w2 + vi * 4 * 256;
  float o0 = 0.f, o1 = 0.f, o2 = 0.f, o3 = 0.f;
#pragma unroll
  for (int j = 0; j < 8; ++j) {
    int c = (j < 4) ? lane : (lane + 32);
    int k = c * 4 + (j & 3);
    float hn = (h[j] - mu) * rstd * glv[k] + blv[k];
    float ge = 0.5f * hn * (1.f + erff(hn * 0.7071067811865475f)); // exact GELU
    o0 += w2v[k] * ge;
    o1 += w2v[256 + k] * ge;
    o2 += w2v[512 + k] * ge;
    o3 += w2v[768 + k] * ge;
  }
  o0 = wave_sum32(o0);
  o1 = wave_sum32(o1);
  o2 = wave_sum32(o2);
  o3 = wave_sum32(o3);
  if (lane < 4) {
    float ov = (lane == 0) ? o0 : (lane == 1) ? o1 : (lane == 2) ? o2 : o3;
    int sc = hq & 1, hk = hq >> 1, wk = wq * 4 + lane;
    float rng = (sc == 0) ? (1.f / 15.f) : (1.f / 127.f);
    float val = tanhf(ov) * rng * OFR_F +
                refp[((size_t)((b * VV + vi) * HK + hk) * WK + wk) * 2 + (1 - sc)];
    rwo[((size_t)(vi * 4 + bg) * NSS + hk * WK + wk) * 2 + sc] = val;
  }
}

// ---------------- bilinear sampling of xs ------------------------------------
__global__ void k_sample(const float* __restrict__ x, const float* __restrict__ rwo,
                         float* __restrict__ xs, int vi) {
  int idx = blockIdx.x * blockDim.x + threadIdx.x;
  int n = idx & (NSS - 1);
  int cq = (idx >> 11) & 15;
  int bg = idx >> 15;
  if (bg >= 4) return;
  int b = bg >> 1, g = bg & 1;
  float r0 = rwo[((size_t)(vi * 4 + bg) * NSS + n) * 2 + 0];
  float r1 = rwo[((size_t)(vi * 4 + bg) * NSS + n) * 2 + 1];
  float xx = (r1 + 1.f) * 0.5f * (WI - 1);
  float yy = (r0 + 1.f) * 0.5f * (HI - 1);
  float x0f = floorf(xx), y0f = floorf(yy);
  int ix0 = (int)x0f, iy0 = (int)y0f, ix1 = ix0 + 1, iy1 = iy0 + 1;
  float wx1 = xx - x0f, wy1 = yy - y0f, wx0 = 1.f - wx1, wy0 = 1.f - wy1;
  float mx0 = (ix0 >= 0 && ix0 < WI) ? 1.f : 0.f;
  float mx1 = (ix1 >= 0 && ix1 < WI) ? 1.f : 0.f;
  float my0 = (iy0 >= 0 && iy0 < HI) ? 1.f : 0.f;
  float my1 = (iy1 >= 0 && iy1 < HI) ? 1.f : 0.f;
  int cx0 = min(max(ix0, 0), WI - 1), cx1 = min(max(ix1, 0), WI - 1);
  int cy0 = min(max(iy0, 0), HI - 1), cy1 = min(max(iy1, 0), HI - 1);
  float w00 = wx0 * wy0 * mx0 * my0, w10 = wx1 * wy0 * mx1 * my0;
  float w01 = wx0 * wy1 * mx0 * my1, w11 = wx1 * wy1 * mx1 * my1;
#pragma unroll
  for (int j = 0; j < 4; ++j) {
    int ch = g * CPG + cq * 4 + j;
    const float* im = x + ((size_t)(b * VV + vi) * CC + ch) * (HI * WI);
    float v = w00 * im[cy0 * WI + cx0] + w10 * im[cy0 * WI + cx1] +
              w01 * im[cy1 * WI + cx0] + w11 * im[cy1 * WI + cx1];
    xs[(size_t)(b * CC + ch) * NSS + n] = v;
  }
}

// ---------------- K/V projection: W(128x128) * xs_b(128x2048) + bias ---------
__global__ void k_proj(const float* __restrict__ W, const float* __restrict__ bias,
                       const float* __restrict__ xs, float* __restrict__ out) {
  int wg = blockIdx.x * (blockDim.x >> 5) + (threadIdx.x >> 5);
  int lane = threadIdx.x & 31;
  int nt = wg & 127;
  int mt = (wg >> 7) & 7;
  int b = wg >> 10;
  if (b >= BB) return;
  int ml = lane & 15, half = lane >> 4, klo = half * 2;
  v8f acc = {};
  int arow = mt * 16 + ml;
  int ncol = nt * 16 + ml;
  const float* xb = xs + (size_t)b * CC * NSS;
#pragma unroll 4
  for (int kc = 0; kc < 32; ++kc) {
    int k0 = kc * 4 + klo;
    v2f a, bm;
    a.x = W[arow * CC + k0];
    a.y = W[arow * CC + k0 + 1];
    bm.x = xb[(size_t)k0 * NSS + ncol];
    bm.y = xb[(size_t)(k0 + 1) * NSS + ncol];
    acc = wmma4(a, bm, acc);
  }
#pragma unroll
  for (int r = 0; r < 8; ++r) {
    int row = mt * 16 + r + 8 * half;
    out[(size_t)(b * CC + row) * NSS + ncol] = acc[r] + bias[row];
  }
}

// ---------------- flash attention: one wave per (head, m-tile) ---------------
__global__ void k_attn(const float* __restrict__ query, const float* __restrict__ kbuf,
                       const float* __restrict__ vbuf, const float* __restrict__ rwo,
                       const float* __restrict__ rpe, float* __restrict__ outv, int vi) {
  __shared__ float lds[4][288];
  int wslot = threadIdx.x >> 5;
  float* P = lds[wslot];        // 256: P^T tile (n-major) for second WMMA B operand
  float* stats = P + 256;       // 16: per-row rescale / final row-sum broadcast
  int wg = blockIdx.x * (blockDim.x >> 5) + wslot;
  int lane = threadIdx.x & 31;
  int mt = wg & 63;
  int bh = wg >> 6;
  if (bh >= BB * NHH) return;
  int b = bh >> 3, hh = bh & 7;
  int g = hh >> 2;
  int bg = b * GG + g;
  int m0 = mt * 16;
  int ml = lane & 15, half = lane >> 4, klo = half * 2;

  v2f qA[4];
#pragma unroll
  for (int kc = 0; kc < 4; ++kc) {
    int ch = kc * 4 + klo;
    qA[kc].x = query[(size_t)(b * CC + hh * CHH + ch) * MQ + m0 + ml];
    qA[kc].y = query[(size_t)(b * CC + hh * CHH + ch + 1) * MQ + m0 + ml];
  }
  float qg0[8], qg1[8];
#pragma unroll
  for (int r = 0; r < 8; ++r) {
    int m = m0 + r + 8 * half;
    qg0[r] = -1.f + 2.f * (float)(m >> 5) * (1.f / 31.f);
    qg1[r] = -1.f + 2.f * (float)(m & 31) * (1.f / 31.f);
  }
  const float* tab = rpe + (size_t)hh * RPE_H * RPE_W;
  const float* rwv = rwo + (size_t)(vi * 4 + bg) * NSS * 2;
  const float* kb = kbuf + (size_t)(b * CC + hh * CHH) * NSS;
  const float* vb = vbuf + (size_t)(b * CC + hh * CHH) * NSS;

  v8f acc = {};
  float rmax[8], rsum[8];
#pragma unroll
  for (int r = 0; r < 8; ++r) { rmax[r] = -1e30f; rsum[r] = 0.f; }

  for (int nt = 0; nt < 128; ++nt) {
    int n0 = nt * 16;
    v2f kB[4];
#pragma unroll
    for (int kc = 0; kc < 4; ++kc) {
      int ch = kc * 4 + klo;
      kB[kc].x = kb[(size_t)ch * NSS + n0 + ml];
      kB[kc].y = kb[(size_t)(ch + 1) * NSS + n0 + ml];
    }
    v8f S = {};
#pragma unroll
    for (int kc = 0; kc < 4; ++kc) S = wmma4(qA[kc], kB[kc], S);

    // rpe bias via bilinear gather; S tile layout: lane -> n, vgpr r -> m row
    float rw0 = rwv[(n0 + ml) * 2 + 0];
    float rw1 = rwv[(n0 + ml) * 2 + 1];
#pragma unroll
    for (int r = 0; r < 8; ++r) {
      float xx = (0.5f * (qg1[r] - rw1) + 1.f) * 0.5f * (RPE_W - 1);
      float yy = (0.5f * (qg0[r] - rw0) + 1.f) * 0.5f * (RPE_H - 1);
      float x0f = floorf(xx), y0f = floorf(yy);
      int ix0 = (int)x0f, iy0 = (int)y0f, ix1 = ix0 + 1, iy1 = iy0 + 1;
      float wx1 = xx - x0f, wy1 = yy - y0f, wx0 = 1.f - wx1, wy0 = 1.f - wy1;
      float mx0 = (ix0 >= 0 && ix0 < RPE_W) ? 1.f : 0.f;
      float mx1 = (ix1 >= 0 && ix1 < RPE_W) ? 1.f : 0.f;
      float my0 = (iy0 >= 0 && iy0 < RPE_H) ? 1.f : 0.f;
      float my1 = (iy1 >= 0 && iy1 < RPE_H) ? 1.f : 0.f;
      int cx0 = min(max(ix0, 0), RPE_W - 1), cx1 = min(max(ix1, 0), RPE_W - 1);
      int cy0 = min(max(iy0, 0), RPE_H - 1), cy1 = min(max(iy1, 0), RPE_H - 1);
      float bias = wx0 * wy0 * mx0 * my0 * tab[cy0 * RPE_W + cx0] +
                   wx1 * wy0 * mx1 * my0 * tab[cy0 * RPE_W + cx1] +
                   wx0 * wy1 * mx0 * my1 * tab[cy1 * RPE_W + cx0] +
                   wx1 * wy1 * mx1 * my1 * tab[cy1 * RPE_W + cx1];
      S[r] = S[r] * SCALE_F + bias;
    }
    // online softmax (row = across 16-lane half)
#pragma unroll
    for (int r = 0; r < 8; ++r) {
      float mx = half_max16(S[r]);
      float nm = fmaxf(rmax[r], mx);
      float sc = __expf(rmax[r] - nm);
      float p = __expf(S[r] - nm);
      rmax[r] = nm;
      rsum[r] = rsum[r] * sc + half_sum16(p);
      if (ml == r) stats[half * 8 + r] = sc;
      P[ml * 16 + r + 8 * half] = p; // store P^T (n-major)
    }
    __syncthreads();
    float scm = stats[ml];
#pragma unroll
    for (int r = 0; r < 8; ++r) acc[r] *= scm;
    // acc(c x m) += V(c x n-chunk) * P^T(n-chunk x m)
#pragma unroll
    for (int kc = 0; kc < 4; ++kc) {
      int kn = kc * 4 + klo;
      v2f vA, pB;
      vA.x = vb[(size_t)ml * NSS + n0 + kn];
      vA.y = vb[(size_t)ml * NSS + n0 + kn + 1];
      pB.x = P[kn * 16 + ml];
      pB.y = P[(kn + 1) * 16 + ml];
      acc = wmma4(vA, pB, acc);
    }
    __syncthreads();
  }
#pragma unroll
  for (int r = 0; r < 8; ++r)
    if (ml == r) stats[half * 8 + r] = rsum[r];
  __syncthreads();
  float inv = 1.f / stats[ml];
#pragma unroll
  for (int r = 0; r < 8; ++r) {
    int cch = hh * CHH + r + 8 * half;
    outv[((size_t)(vi * BB + b) * CC + cch) * MQ + m0 + ml] = acc[r] * inv;
  }
}

// ---------------- final projection: out_w(128x384) * outv(384x1024) + bias ---
__global__ void k_outproj(const float* __restrict__ Wo, const float* __restrict__ bo,
                          const float* __restrict__ outv, float* __restrict__ out) {
  int wg = blockIdx.x * (blockDim.x >> 5) + (threadIdx.x >> 5);
  int lane = threadIdx.x & 31;
  int nt = wg & 63;
  int mt = (wg >> 6) & 7;
  int b = wg >> 9;
  if (b >= BB) return;
  int ml = lane & 15, half = lane >> 4, klo = half * 2;
  v8f acc = {};
  int arow = mt * 16 + ml;
  int ncol = nt * 16 + ml;
#pragma unroll 4
  for (int kc = 0; kc < 96; ++kc) {
    int k0 = kc * 4 + klo;
    v2f a, bm;
    a.x = Wo[arow * (CC * VV) + k0];
    a.y = Wo[arow * (CC * VV) + k0 + 1];
    int v0 = k0 >> 7, c0 = k0 & 127; // k0 even -> k0,k0+1 share the same view
    bm.x = outv[((size_t)(v0 * BB + b) * CC + c0) * MQ + ncol];
    bm.y = outv[((size_t)(v0 * BB + b) * CC + c0 + 1) * MQ + ncol];
    acc = wmma4(a, bm, acc);
  }
#pragma unroll
  for (int r = 0; r < 8; ++r) {
    int row = mt * 16 + r + 8 * half;
    out[(size_t)(b * CC + row) * MQ + ncol] = acc[r] + bo[row];
  }
}

extern "C" void kernel_launch(void* const* d_in, const int* in_sizes, int n_in,
                              void* d_out, int out_size, void* d_ws, size_t ws_size,
                              hipStream_t stream) {
  const float* x = (const float*)d_in[0];
  const float* query = (const float*)d_in[1];
  const float* refp = (const float*)d_in[2];
  const float* off_w1 = (const float*)d_in[3];
  const float* off_b1 = (const float*)d_in[4];
  const float* off_ln_g = (const float*)d_in[5];
  const float* off_ln_b = (const float*)d_in[6];
  const float* off_w2 = (const float*)d_in[7];
  const float* k_w = (const float*)d_in[8];
  const float* k_b = (const float*)d_in[9];
  const float* v_w = (const float*)d_in[10];
  const float* v_b = (const float*)d_in[11];
  const float* out_w = (const float*)d_in[12];
  const float* out_b = (const float*)d_in[13];
  const float* rpe = (const float*)d_in[14];

  float* rwo = (float*)d_ws;                       // V*4*NS*2   = 49152 f
  float* xs = rwo + (size_t)VV * 4 * NSS * 2;      // B*C*NS     = 524288 f
  float* kbuf = xs + (size_t)BB * CC * NSS;        // B*C*NS
  float* vbuf = kbuf + (size_t)BB * CC * NSS;      // B*C*NS
  float* outv = vbuf + (size_t)BB * CC * NSS;      // V*B*C*1024 = 786432 f

  k_offset<<<1536, 256, 0, stream>>>(query, refp, off_w1, off_b1, off_ln_g, off_ln_b,
                                     off_w2, rwo);
  for (int vi = 0; vi < VV; ++vi) {
    k_sample<<<512, 256, 0, stream>>>(x, rwo, xs, vi);
    k_proj<<<512, 128, 0, stream>>>(k_w, k_b, xs, kbuf);
    k_proj<<<512, 128, 0, stream>>>(v_w, v_b, xs, vbuf);
    k_attn<<<256, 128, 0, stream>>>(query, kbuf, vbuf, rwo, rpe, outv, vi);
  }
  k_outproj<<<256, 128, 0, stream>>>(out_w, out_b, outv, (float*)d_out);
}